// MultiHeadAttention_48679159332956
// MI455X (gfx1250) — compile-verified
//
#include <hip/hip_runtime.h>
#include <math.h>

#define D_MODEL 1024
#define N_HEADS 16
#define D_K     64
#define SEQ     2048
#define BATCH   2

typedef float v2f __attribute__((ext_vector_type(2)));
typedef float v8f __attribute__((ext_vector_type(8)));

// D = A(16x4,f32) * B(4x16,f32) + C(16x16,f32)  -> v_wmma_f32_16x16x4_f32
__device__ __forceinline__ v8f wmma_f32(v2f a, v2f b, v8f c) {
  return __builtin_amdgcn_wmma_f32_16x16x4_f32(
      /*neg_a=*/false, a, /*neg_b=*/false, b,
      /*c_mod=*/(short)0, c, /*reuse_a=*/false, /*reuse_b=*/false);
}

__device__ __forceinline__ v8f v8zero() {
  v8f z;
#pragma unroll
  for (int i = 0; i < 8; ++i) z[i] = 0.0f;
  return z;
}

// A fragment (16x4) from row-major LDS: lane m = l&15; K = k0 + 2*(l>>4) + {0,1}
__device__ __forceinline__ v2f frag_a(const float* p, int ld, int row0, int k0, int lane) {
  const float* q = p + (row0 + (lane & 15)) * ld + k0 + ((lane >> 4) << 1);
  v2f a; a.x = q[0]; a.y = q[1]; return a;
}
// B fragment (4x16) from row-major [K x ld] LDS: lane n = l&15; K rows k0+2*(l>>4)+{0,1}
__device__ __forceinline__ v2f frag_b(const float* p, int ld, int k0, int n0, int lane) {
  int kb = k0 + ((lane >> 4) << 1);
  int n  = n0 + (lane & 15);
  v2f b; b.x = p[kb * ld + n]; b.y = p[(kb + 1) * ld + n]; return b;
}
// B fragment where memory is [n x ld] and we need B[k][n] = mem[n][k] (Q*K^T path)
__device__ __forceinline__ v2f frag_bt(const float* p, int ld, int n0, int k0, int lane) {
  const float* q = p + (n0 + (lane & 15)) * ld + k0 + ((lane >> 4) << 1);
  v2f b; b.x = q[0]; b.y = q[1]; return b;
}

// ---------------------------------------------------------------------------
// C[m,n] = sum_k A[m,k] * W[k,n] + bias[n]
// Tile 64x64 per 256-thread WG (8 waves: wave -> (mTile = w&3, nHalf = w>>2)).
// headSplit=1 writes [b, h, s, dk] layout for the attention kernel.
// ---------------------------------------------------------------------------
__global__ __launch_bounds__(256) void gemm_bias_64x64(
    const float* __restrict__ A, const float* __restrict__ W,
    const float* __restrict__ bias, float* __restrict__ C, int headSplit)
{
  __shared__ __align__(16) float lda[64 * 20];  // 64x16, stride 20 (bank/align safe)
  __shared__ __align__(16) float ldw[16 * 68];  // 16x64, stride 68

  const int t = threadIdx.x, lane = t & 31, wave = t >> 5;
  const int m0 = blockIdx.x * 64, n0 = blockIdx.y * 64;
  const int wm = (wave & 3) << 4;   // wave row offset within tile
  const int wn = (wave >> 2) << 5;  // wave col offset (0 or 32); covers 2 n-subtiles
  const int nl = lane & 15, hi = lane >> 4;

  float bb0 = bias[n0 + wn + nl];
  float bb1 = bias[n0 + wn + 16 + nl];
  v8f c0, c1;
#pragma unroll
  for (int r = 0; r < 8; ++r) { c0[r] = bb0; c1[r] = bb1; }

  const int arow = t >> 2, ac4 = (t & 3) << 2;   // stage A: 64x16
  const int wrow = t >> 4, wc4 = (t & 15) << 2;  // stage W: 16x64

  for (int k0 = 0; k0 < D_MODEL; k0 += 16) {
    float4 av = *(const float4*)(A + (size_t)(m0 + arow) * D_MODEL + k0 + ac4);
    float4 wv = *(const float4*)(W + (size_t)(k0 + wrow) * D_MODEL + n0 + wc4);
    *(float4*)&lda[arow * 20 + ac4] = av;
    *(float4*)&ldw[wrow * 68 + wc4] = wv;
    if (k0 + 16 < D_MODEL) {  // warm next K-slab (global_prefetch_b8)
      __builtin_prefetch(A + (size_t)(m0 + arow) * D_MODEL + k0 + 16 + ac4, 0, 3);
      __builtin_prefetch(W + (size_t)(k0 + 16 + wrow) * D_MODEL + n0 + wc4, 0, 3);
    }
    __syncthreads();
#pragma unroll
    for (int kk = 0; kk < 16; kk += 4) {
      v2f a  = frag_a(lda, 20, wm, kk, lane);
      v2f b0 = frag_b(ldw, 68, kk, wn, lane);
      v2f b1 = frag_b(ldw, 68, kk, wn + 16, lane);
      c0 = wmma_f32(a, b0, c0);
      c1 = wmma_f32(a, b1, c1);
    }
    __syncthreads();
  }

#pragma unroll
  for (int r = 0; r < 8; ++r) {
    int m  = m0 + wm + r + (hi << 3);
    int cn = n0 + wn + nl;
    if (headSplit) {
      int b = m >> 11, s = m & (SEQ - 1);
      int cn1 = cn + 16;
      size_t i0 = ((((size_t)b * N_HEADS) + (cn  >> 6)) * SEQ + s) * D_K + (cn  & 63);
      size_t i1 = ((((size_t)b * N_HEADS) + (cn1 >> 6)) * SEQ + s) * D_K + (cn1 & 63);
      C[i0] = c0[r];
      C[i1] = c1[r];
    } else {
      C[(size_t)m * D_MODEL + cn]      = c0[r];
      C[(size_t)m * D_MODEL + cn + 16] = c1[r];
    }
  }
}

// ---------------------------------------------------------------------------
// Fused attention for a 64-row q-tile of one (b,h):
//  phase 1: row max of scores   (WMMA recompute, no storage)
//  phase 2: row sum of exp      (WMMA recompute)
//  phase 3: write normalized weights (single 537MB stream) + ctx = W @ V (WMMA)
// wave -> (mt = w&3 : 16 q-rows, nt = w>>2 : which of 2 key sub-tiles)
// ---------------------------------------------------------------------------
__global__ __launch_bounds__(256) void mha_attention_64q(
    const float* __restrict__ qh, const float* __restrict__ kh,
    const float* __restrict__ vh, float* __restrict__ attn,
    float* __restrict__ ctx)
{
  __shared__ __align__(16) float lq[64 * 68];    // Q tile (pre-scaled)
  __shared__ __align__(16) float lk[32 * 68];    // 2 key tiles
  __shared__ __align__(16) float lv[32 * 68];    // 2 value tiles
  __shared__ __align__(16) float lw[8 * 16 * 20];// per-wave 16x16 weight staging
  __shared__ __align__(16) float lctx[64 * 68];  // cross-wave ctx reduction
  __shared__ float pmax[64 * 2];                 // [row][nt] partial max (single writer)
  __shared__ float psum[64 * 2];                 // [row][nt] partial sum (single writer)
  __shared__ float rmax[64];
  __shared__ float rsuminv[64];

  const int t = threadIdx.x, lane = t & 31, wave = t >> 5;
  const int mt = wave & 3, nt = wave >> 2;
  const int nl = lane & 15, hi = lane >> 4;
  const int qt = blockIdx.x, h = blockIdx.y, b = blockIdx.z;

  const float* qp = qh + ((((size_t)b * N_HEADS + h) * SEQ) + qt * 64) * D_K;
  const float* kp = kh + (((size_t)b * N_HEADS + h) * SEQ) * D_K;
  const float* vp = vh + (((size_t)b * N_HEADS + h) * SEQ) * D_K;
  float* ap = attn + ((((size_t)b * N_HEADS + h) * SEQ) + qt * 64) * SEQ;
  float* cp = ctx + (((size_t)b * SEQ) + qt * 64) * D_MODEL + h * D_K;

  // Q tile, pre-scaled by 1/sqrt(d_k) = 0.125
  for (int i = t; i < 64 * 16; i += 256) {
    int row = i >> 4, c4 = (i & 15) << 2;
    float4 v4 = *(const float4*)(qp + (size_t)row * D_K + c4);
    v4.x *= 0.125f; v4.y *= 0.125f; v4.z *= 0.125f; v4.w *= 0.125f;
    *(float4*)&lq[row * 68 + c4] = v4;
  }
  if (t < 128) { pmax[t] = -INFINITY; psum[t] = 0.0f; }
  __syncthreads();

  // ---- phase 1: row max ----
  for (int kt = 0; kt < SEQ / 32; ++kt) {
    for (int i = t; i < 32 * 16; i += 256) {
      int row = i >> 4, c4 = (i & 15) << 2;
      *(float4*)&lk[row * 68 + c4] =
          *(const float4*)(kp + (size_t)(kt * 32 + row) * D_K + c4);
    }
    __syncthreads();
    v8f sc = v8zero();
#pragma unroll
    for (int kk = 0; kk < D_K; kk += 4) {
      v2f a  = frag_a(lq, 68, mt * 16, kk, lane);
      v2f bf = frag_bt(lk, 68, nt * 16, kk, lane);
      sc = wmma_f32(a, bf, sc);
    }
#pragma unroll
    for (int r = 0; r < 8; ++r) {
      float mv = sc[r];
      mv = fmaxf(mv, __shfl_xor(mv, 1, 32));
      mv = fmaxf(mv, __shfl_xor(mv, 2, 32));
      mv = fmaxf(mv, __shfl_xor(mv, 4, 32));
      mv = fmaxf(mv, __shfl_xor(mv, 8, 32));
      if (nl == 0) {
        int row = mt * 16 + r + (hi << 3);
        pmax[row * 2 + nt] = fmaxf(pmax[row * 2 + nt], mv);
      }
    }
    __syncthreads();
  }
  if (t < 64) rmax[t] = fmaxf(pmax[t * 2], pmax[t * 2 + 1]);
  __syncthreads();

  // ---- phase 2: row sum of exp ----
  for (int kt = 0; kt < SEQ / 32; ++kt) {
    for (int i = t; i < 32 * 16; i += 256) {
      int row = i >> 4, c4 = (i & 15) << 2;
      *(float4*)&lk[row * 68 + c4] =
          *(const float4*)(kp + (size_t)(kt * 32 + row) * D_K + c4);
    }
    __syncthreads();
    v8f sc = v8zero();
#pragma unroll
    for (int kk = 0; kk < D_K; kk += 4) {
      v2f a  = frag_a(lq, 68, mt * 16, kk, lane);
      v2f bf = frag_bt(lk, 68, nt * 16, kk, lane);
      sc = wmma_f32(a, bf, sc);
    }
#pragma unroll
    for (int r = 0; r < 8; ++r) {
      int row = mt * 16 + r + (hi << 3);
      float e = __expf(sc[r] - rmax[row]);
      e += __shfl_xor(e, 1, 32);
      e += __shfl_xor(e, 2, 32);
      e += __shfl_xor(e, 4, 32);
      e += __shfl_xor(e, 8, 32);
      if (nl == 0) psum[row * 2 + nt] += e;
    }
    __syncthreads();
  }
  if (t < 64) rsuminv[t] = 1.0f / (psum[t * 2] + psum[t * 2 + 1]);
  __syncthreads();

  // ---- phase 3: write normalized weights + accumulate ctx = W @ V ----
  v8f o0 = v8zero(), o1 = v8zero(), o2 = v8zero(), o3 = v8zero();
  float* lwp = lw + wave * 320;  // private 16x20 staging
  for (int kt = 0; kt < SEQ / 32; ++kt) {
    for (int i = t; i < 32 * 16; i += 256) {
      int row = i >> 4, c4 = (i & 15) << 2;
      *(float4*)&lk[row * 68 + c4] =
          *(const float4*)(kp + (size_t)(kt * 32 + row) * D_K + c4);
      *(float4*)&lv[row * 68 + c4] =
          *(const float4*)(vp + (size_t)(kt * 32 + row) * D_K + c4);
    }
    __syncthreads();
    v8f sc = v8zero();
#pragma unroll
    for (int kk = 0; kk < D_K; kk += 4) {
      v2f a  = frag_a(lq, 68, mt * 16, kk, lane);
      v2f bf = frag_bt(lk, 68, nt * 16, kk, lane);
      sc = wmma_f32(a, bf, sc);
    }
#pragma unroll
    for (int r = 0; r < 8; ++r) {
      int row = mt * 16 + r + (hi << 3);
      float wv = __expf(sc[r] - rmax[row]) * rsuminv[row];
      ap[(size_t)row * SEQ + kt * 32 + nt * 16 + nl] = wv;  // coalesced stream
      lwp[(r + (hi << 3)) * 20 + nl] = wv;                  // A-major restage
    }
#pragma unroll
    for (int kk = 0; kk < 16; kk += 4) {
      v2f a = frag_a(lwp, 20, 0, kk, lane);
      o0 = wmma_f32(a, frag_b(lv, 68, nt * 16 + kk, 0,  lane), o0);
      o1 = wmma_f32(a, frag_b(lv, 68, nt * 16 + kk, 16, lane), o1);
      o2 = wmma_f32(a, frag_b(lv, 68, nt * 16 + kk, 32, lane), o2);
      o3 = wmma_f32(a, frag_b(lv, 68, nt * 16 + kk, 48, lane), o3);
    }
    __syncthreads();
  }

  // cross-wave reduce the two key-halves, write ctx in [b, s, h*64+dk]
  if (nt == 1) {
#pragma unroll
    for (int r = 0; r < 8; ++r) {
      int row = mt * 16 + r + (hi << 3);
      lctx[row * 68 + 0  + nl] = o0[r];
      lctx[row * 68 + 16 + nl] = o1[r];
      lctx[row * 68 + 32 + nl] = o2[r];
      lctx[row * 68 + 48 + nl] = o3[r];
    }
  }
  __syncthreads();
  if (nt == 0) {
#pragma unroll
    for (int r = 0; r < 8; ++r) {
      int row = mt * 16 + r + (hi << 3);
      float* dst = cp + (size_t)row * D_MODEL;
      dst[0  + nl] = o0[r] + lctx[row * 68 + 0  + nl];
      dst[16 + nl] = o1[r] + lctx[row * 68 + 16 + nl];
      dst[32 + nl] = o2[r] + lctx[row * 68 + 32 + nl];
      dst[48 + nl] = o3[r] + lctx[row * 68 + 48 + nl];
    }
  }
}

// ---------------------------------------------------------------------------
extern "C" void kernel_launch(void* const* d_in, const int* in_sizes, int n_in,
                              void* d_out, int out_size, void* d_ws, size_t ws_size,
                              hipStream_t stream) {
  (void)in_sizes; (void)n_in; (void)out_size; (void)ws_size;
  const float* Q  = (const float*)d_in[0];
  const float* K  = (const float*)d_in[1];
  const float* V  = (const float*)d_in[2];
  const float* Wq = (const float*)d_in[3];
  const float* bq = (const float*)d_in[4];
  const float* Wk = (const float*)d_in[5];
  const float* bk = (const float*)d_in[6];
  const float* Wv = (const float*)d_in[7];
  const float* bv = (const float*)d_in[8];
  const float* Wo = (const float*)d_in[9];
  const float* bo = (const float*)d_in[10];

  float* out  = (float*)d_out;
  float* attn = out + (size_t)BATCH * SEQ * D_MODEL;  // tuple order: (output, attn)

  const size_t elems = (size_t)BATCH * SEQ * D_MODEL; // 4,194,304 per tensor
  float* qhp = (float*)d_ws;        // [b,h,s,dk]
  float* khp = qhp + elems;
  float* vhp = khp + elems;
  float* ctx = vhp + elems;         // [b,s,d_model]

  dim3 blk(256);
  dim3 gg(BATCH * SEQ / 64, D_MODEL / 64);  // 64 x 16 tiles

  gemm_bias_64x64<<<gg, blk, 0, stream>>>(Q, Wq, bq, qhp, 1);
  gemm_bias_64x64<<<gg, blk, 0, stream>>>(K, Wk, bk, khp, 1);
  gemm_bias_64x64<<<gg, blk, 0, stream>>>(V, Wv, bv, vhp, 1);
  mha_attention_64q<<<dim3(SEQ / 64, N_HEADS, BATCH), blk, 0, stream>>>(
      qhp, khp, vhp, attn, ctx);
  gemm_bias_64x64<<<gg, blk, 0, stream>>>(ctx, Wo, bo, out, 0);
}